// TextSceneAttentionFusion_67654324847585
// MI455X (gfx1250) — compile-verified
//
#include <hip/hip_runtime.h>
#include <math.h>

typedef float v2f __attribute__((ext_vector_type(2)));
typedef float v8f __attribute__((ext_vector_type(8)));

#define S_TOTAL 32768   // D*H*Wd = 32*32*32
#define SFEAT   256
#define B_DIM   2

// ws layout (floats):
//   [0, 65536)        c[b*S + s]
//   [65536, 65538)    m[b]
//   [65538, 65540)    Z[b]
//   [65544, 66056)    F[b*256 + f]

// ---------------- Pass 1: c[b,s] = scene[b,s,:] . Ws  via WMMA f32 16x16x4 ----
// A (16x4, documented layout): lane L holds A[L&15, 2*(L>>4)] , A[L&15, 2*(L>>4)+1]
// B (4x16): all 16 columns = Ws chunk -> lane L holds Ws[f0+2*(L>>4)+{0,1}]
// D: every column identical = c[row0+m]; lane L, vgpr g = D[g + 8*(L>>4), L&15]
__global__ __launch_bounds__(256) void ts_fuse_gemv_kernel(
    const float* __restrict__ scene, const float* __restrict__ W,
    float* __restrict__ cbuf) {
  const float* Ws = W + 256;                 // TF = 256
  const int tid  = threadIdx.x;
  const int lane = tid & 31;
  const int wave = tid >> 5;                 // 0..7
  const int hi   = lane >> 4;                // 0/1 lane half
  const int mrow = lane & 15;
  const int b    = blockIdx.x >> 6;          // grid = 2 * 64
  const int blk  = blockIdx.x & 63;
  const long long base = (long long)b * S_TOTAL + blk * 512 + wave * 64;

  for (int chunk = 0; chunk < 4; ++chunk) {
    const long long row0 = base + chunk * 16;
    const float* arow = scene + (row0 + mrow) * SFEAT + 2 * hi;
    v8f acc = {};
    #pragma unroll 8
    for (int f0 = 0; f0 < SFEAT; f0 += 4) {
      v2f a  = *(const v2f*)(arow + f0);               // 8B aligned (even idx)
      v2f bv = *(const v2f*)(Ws + f0 + 2 * hi);        // replicated columns
      acc = __builtin_amdgcn_wmma_f32_16x16x4_f32(
          false, a, false, bv, (short)0, acc, false, false);
    }
    if (mrow == 0) {                                   // lanes 0 and 16
      float* cw = cbuf + row0 + 8 * hi;
      #pragma unroll
      for (int g = 0; g < 8; ++g) cw[g] = acc[g];
    }
  }
}

// ---------------- Pass 2: softmax stats m[b], Z[b]; zero F -------------------
__global__ __launch_bounds__(256) void ts_fuse_stats_kernel(
    const float* __restrict__ cbuf, float* __restrict__ stats,
    float* __restrict__ F) {
  __shared__ float red[256];
  const int b = blockIdx.x;
  const int t = threadIdx.x;
  const float* c = cbuf + (long long)b * S_TOTAL;

  float mloc = -INFINITY;
  for (int k = t; k < S_TOTAL; k += 256) mloc = fmaxf(mloc, c[k]);
  red[t] = mloc; __syncthreads();
  for (int s = 128; s > 0; s >>= 1) {
    if (t < s) red[t] = fmaxf(red[t], red[t + s]);
    __syncthreads();
  }
  const float m = red[0];
  __syncthreads();

  float zloc = 0.0f;
  for (int k = t; k < S_TOTAL; k += 256) zloc += __expf(c[k] - m);
  red[t] = zloc; __syncthreads();
  for (int s = 128; s > 0; s >>= 1) {
    if (t < s) red[t] += red[t + s];
    __syncthreads();
  }
  if (t == 0) { stats[b] = m; stats[2 + b] = red[0]; }
  F[b * 256 + t] = 0.0f;
}

// ---------------- Pass 3: F[b,f] += sum_s exp(c-m) * scene[b,s,f] via WMMA ---
// A (16x4): A[m,k] = scene[s0+k, ftile+m]  (M = feature, K = s-rows)
// B (4x16): all columns = exp(c[s0+k]-m)   (replicated -> only K mapping used)
// D: every column = partial F[ftile+m]; lane L, vgpr g = D[g+8*(L>>4), L&15]
__global__ __launch_bounds__(512) void ts_fuse_wsum_kernel(
    const float* __restrict__ scene, const float* __restrict__ cbuf,
    const float* __restrict__ stats, float* __restrict__ F) {
  const int tid  = threadIdx.x;
  const int lane = tid & 31;
  const int wave = tid >> 5;                 // 0..15 -> f-tile
  const int hi   = lane >> 4;
  const int n    = lane & 15;
  const int b    = blockIdx.x >> 5;          // grid = 2 * 32
  const int blk  = blockIdx.x & 31;
  const float m  = stats[b];
  const long long rowBase = (long long)b * S_TOTAL + blk * 1024;
  const int ftile = wave * 16;

  v8f acc = {};
  for (int i = 0; i < 256; ++i) {            // 1024 s-rows, 4 per WMMA
    const long long r0 = rowBase + i * 4 + 2 * hi;
    const float* p0 = scene + r0 * SFEAT + ftile + n;
    v2f a;  a.x = p0[0];  a.y = p0[SFEAT];   // rows r0, r0+1
    v2f cp = *(const v2f*)(cbuf + r0);       // even index -> 8B aligned
    v2f bv; bv.x = __expf(cp.x - m); bv.y = __expf(cp.y - m);
    acc = __builtin_amdgcn_wmma_f32_16x16x4_f32(
        false, a, false, bv, (short)0, acc, false, false);
  }
  if (n == 0) {                              // lanes 0 and 16 hold F partials
    float* Fp = F + b * 256 + ftile + 8 * hi;
    #pragma unroll
    for (int g = 0; g < 8; ++g) atomicAdd(&Fp[g], acc[g]);
  }
}

// ---------------- Pass 4: out[b,t,f] = F[b,f] / Z[b]  (broadcast over t) -----
__global__ __launch_bounds__(256) void ts_fuse_bcast_kernel(
    const float* __restrict__ F, const float* __restrict__ stats,
    float* __restrict__ out) {
  const int i = blockIdx.x * 256 + threadIdx.x;   // 0 .. 131071
  const int f = i & 255;
  const int b = i >> 16;                          // T*TF = 65536 per batch
  out[i] = F[b * 256 + f] * (1.0f / stats[2 + b]);
}

extern "C" void kernel_launch(void* const* d_in, const int* in_sizes, int n_in,
                              void* d_out, int out_size, void* d_ws, size_t ws_size,
                              hipStream_t stream) {
  // inputs: text(2,256,256) [cancels in softmax], scene(2,32,32,32,256),
  //         W(512), b(1) [cancels in softmax]
  const float* scene = (const float*)d_in[1];
  const float* W     = (const float*)d_in[2];
  float* ws    = (float*)d_ws;
  float* cbuf  = ws;            // 65536 floats
  float* stats = ws + 65536;    // m[2], Z[2]
  float* F     = ws + 65544;    // 512 floats
  float* out   = (float*)d_out;

  ts_fuse_gemv_kernel <<<128, 256, 0, stream>>>(scene, W, cbuf);
  ts_fuse_stats_kernel<<<2,   256, 0, stream>>>(cbuf, stats, F);
  ts_fuse_wsum_kernel <<<64,  512, 0, stream>>>(scene, cbuf, stats, F);
  ts_fuse_bcast_kernel<<<512, 256, 0, stream>>>(F, stats, out);

  (void)d_in; (void)in_sizes; (void)n_in; (void)out_size; (void)ws_size;
}